// GCN_27410481283413
// MI455X (gfx1250) — compile-verified
//
#include <hip/hip_runtime.h>

// Problem constants (from reference): N=50000, K=20, V_FEATS=FILTERS=128.
#define NN 50000
#define KK 20
#define FF 128
#define STRIPS (NN / 16)   // 3125 exactly (50000 % 16 == 0)

typedef float v2f __attribute__((ext_vector_type(2)));
typedef float v8f __attribute__((ext_vector_type(8)));

// ---------------------------------------------------------------------------
// Kernel A: fp32 WMMA projection GEMM.  P = vertices @ W for 3 weight mats.
// One wave computes a 16-row strip of all 128 output columns (8 col tiles),
// accumulating over K=128 in steps of 4 via V_WMMA_F32_16X16X4_F32.
//
// ISA layouts (CDNA5 ISA §7.12.2):
//   A 16x4 f32 : lane l<16 -> row=l, {v0,v1}={K0,K1}; lane>=16 -> {K2,K3}
//   B 4x16 f32 : lane l<16 -> col=l, {v0,v1}={K0,K1}; lane>=16 -> {K2,K3}
//   C/D 16x16  : vgpr j, lane<16 -> M=j,N=lane ; lane>=16 -> M=j+8,N=lane-16
// ---------------------------------------------------------------------------
__global__ __launch_bounds__(256) void gcn_proj_wmma(
    const float* __restrict__ V,
    const float* __restrict__ Wvc,
    const float* __restrict__ Wint,
    const float* __restrict__ Wnh,
    float* __restrict__ Zc,     // d_out z-region
    float* __restrict__ Pint,   // ws
    float* __restrict__ Pnh)    // ws
{
    const int lane  = threadIdx.x & 31;
    const int strip = blockIdx.x * 8 + (threadIdx.x >> 5);
    if (strip >= STRIPS) return;                 // wave-uniform: EXEC stays all-1s
    const int m = blockIdx.y;                    // which weight matrix
    const float* __restrict__ W = (m == 0) ? Wvc : (m == 1) ? Wint : Wnh;
    float*       __restrict__ O = (m == 0) ? Zc  : (m == 1) ? Pint : Pnh;

    const int half = lane >> 4;                  // 0 or 1
    const int l15  = lane & 15;
    const int row0 = strip * 16;

    const float* __restrict__ aBase = V + (size_t)(row0 + l15) * FF + half * 2;

    v8f acc[8];
    const v8f vzero = {0.f, 0.f, 0.f, 0.f, 0.f, 0.f, 0.f, 0.f};
#pragma unroll
    for (int c = 0; c < 8; ++c) acc[c] = vzero;

#pragma unroll 4
    for (int k0 = 0; k0 < FF; k0 += 4) {
        v2f a;
        a.x = aBase[k0 + 0];
        a.y = aBase[k0 + 1];
        const float* __restrict__ bBase = W + (size_t)(k0 + half * 2) * FF + l15;
#pragma unroll
        for (int c = 0; c < 8; ++c) {
            v2f b;
            b.x = bBase[c * 16];
            b.y = bBase[c * 16 + FF];
            // 8 args: (neg_a, A, neg_b, B, c_mod, C, reuse_a, reuse_b)
            acc[c] = __builtin_amdgcn_wmma_f32_16x16x4_f32(
                false, a, false, b, (short)0, acc[c], false, false);
        }
    }

    const int mBase = half * 8;
#pragma unroll
    for (int c = 0; c < 8; ++c) {
#pragma unroll
        for (int j = 0; j < 8; ++j) {
            O[(size_t)(row0 + mBase + j) * FF + c * 16 + l15] = acc[c][j];
        }
    }
}

// ---------------------------------------------------------------------------
// Kernel B: masked gather-aggregate (+bias, +ReLU), bandwidth-bound phase.
// One wave per node; lane l owns features [4l, 4l+4) -> each gathered row is
// one coalesced 512B float4 load across the wave.  Indices/edges are wave-
// uniform -> scalarized with readfirstlane; the mask branch is uniform.
// Reads Zc from the z-region of d_out and overwrites it with the result.
// ---------------------------------------------------------------------------
__global__ __launch_bounds__(256) void gcn_agg(
    const float* __restrict__ Pint,
    const float* __restrict__ Pnh,
    const int*   __restrict__ nhIdx,
    const int*   __restrict__ intIdx,
    const float* __restrict__ nhE,
    const float* __restrict__ intE,
    const float* __restrict__ bv,
    float*       __restrict__ z)
{
    const int lane = threadIdx.x & 31;
    const int node = blockIdx.x * 8 + (threadIdx.x >> 5);
    if (node >= NN) return;
    const int fb = lane * 4;

    const int*   __restrict__ nIp = nhIdx  + (size_t)node * KK;
    const int*   __restrict__ iIp = intIdx + (size_t)node * KK;
    const float* __restrict__ nEp = nhE    + (size_t)node * KK;
    const float* __restrict__ iEp = intE   + (size_t)node * KK;

    float4 aN = make_float4(0.f, 0.f, 0.f, 0.f);
    float4 aI = make_float4(0.f, 0.f, 0.f, 0.f);
    int cN = 0, cI = 0;

    for (int k = 0; k < KK; ++k) {
        const int in = __builtin_amdgcn_readfirstlane(nIp[k]);
        const int ii = __builtin_amdgcn_readfirstlane(iIp[k]);
        if (in >= 0) {                                   // wave-uniform branch
            const float  e = nEp[k];
            const float4 g = *(const float4*)(Pnh + (size_t)in * FF + fb);
            aN.x = fmaf(e, g.x, aN.x);
            aN.y = fmaf(e, g.y, aN.y);
            aN.z = fmaf(e, g.z, aN.z);
            aN.w = fmaf(e, g.w, aN.w);
            ++cN;
        }
        if (ii >= 0) {                                   // wave-uniform branch
            const float  e = iEp[k];
            const float4 g = *(const float4*)(Pint + (size_t)ii * FF + fb);
            aI.x = fmaf(e, g.x, aI.x);
            aI.y = fmaf(e, g.y, aI.y);
            aI.z = fmaf(e, g.z, aI.z);
            aI.w = fmaf(e, g.w, aI.w);
            ++cI;
        }
    }

    const float rN = 1.0f / (float)(cN > 0 ? cN : 1);
    const float rI = 1.0f / (float)(cI > 0 ? cI : 1);

    const float4 zc = *(const float4*)(z  + (size_t)node * FF + fb);
    const float4 b4 = *(const float4*)(bv + fb);

    float4 r;
    r.x = fmaxf(zc.x + aN.x * rN + aI.x * rI + b4.x, 0.f);
    r.y = fmaxf(zc.y + aN.y * rN + aI.y * rI + b4.y, 0.f);
    r.z = fmaxf(zc.z + aN.z * rN + aI.z * rI + b4.z, 0.f);
    r.w = fmaxf(zc.w + aN.w * rN + aI.w * rI + b4.w, 0.f);

    *(float4*)(z + (size_t)node * FF + fb) = r;
}

// ---------------------------------------------------------------------------
// kernel_launch
// d_in : [0]=vertices [1]=nh_indices [2]=int_indices [3]=nh_edges
//        [4]=int_edges [5]=is_int [6]=Wvc [7]=Wvn_int [8]=Wvn_nh [9]=bv
// d_out: z (N*F f32) | nh_indices | int_indices | nh_edges | int_edges | is_int
// d_ws : Pint (N*F f32) | Pnh (N*F f32)   -> needs 51.2 MB
// ---------------------------------------------------------------------------
extern "C" void kernel_launch(void* const* d_in, const int* in_sizes, int n_in,
                              void* d_out, int out_size, void* d_ws, size_t ws_size,
                              hipStream_t stream) {
    const float* vertices = (const float*)d_in[0];
    const int*   nh_idx   = (const int*)  d_in[1];
    const int*   int_idx  = (const int*)  d_in[2];
    const float* nh_e     = (const float*)d_in[3];
    const float* int_e    = (const float*)d_in[4];
    const float* Wvc      = (const float*)d_in[6];
    const float* Wvn_int  = (const float*)d_in[7];
    const float* Wvn_nh   = (const float*)d_in[8];
    const float* bv       = (const float*)d_in[9];

    float* z    = (float*)d_out;
    float* Pint = (float*)d_ws;
    float* Pnh  = Pint + (size_t)NN * FF;

    // Phase 1: the three projections (Zc -> d_out, Pint/Pnh -> ws).
    dim3 gridA((STRIPS + 7) / 8, 3);
    gcn_proj_wmma<<<gridA, dim3(256), 0, stream>>>(
        vertices, Wvc, Wvn_int, Wvn_nh, z, Pint, Pnh);

    // Phase 2: gather/aggregate + bias + ReLU, in-place on z.
    dim3 gridB((NN + 7) / 8);
    gcn_agg<<<gridB, dim3(256), 0, stream>>>(
        Pint, Pnh, nh_idx, int_idx, nh_e, int_e, bv, z);

    // Phase 3: passthrough tuple members (bit-exact d2d copies).
    char* tail = (char*)d_out + (size_t)NN * FF * sizeof(float);
    hipMemcpyAsync(tail, d_in[1], (size_t)NN * KK * 4, hipMemcpyDeviceToDevice, stream);
    tail += (size_t)NN * KK * 4;
    hipMemcpyAsync(tail, d_in[2], (size_t)NN * KK * 4, hipMemcpyDeviceToDevice, stream);
    tail += (size_t)NN * KK * 4;
    hipMemcpyAsync(tail, d_in[3], (size_t)NN * KK * 4, hipMemcpyDeviceToDevice, stream);
    tail += (size_t)NN * KK * 4;
    hipMemcpyAsync(tail, d_in[4], (size_t)NN * KK * 4, hipMemcpyDeviceToDevice, stream);
    tail += (size_t)NN * KK * 4;
    hipMemcpyAsync(tail, d_in[5], (size_t)NN * 4, hipMemcpyDeviceToDevice, stream);
}